// VSSPricerCOSTorch_30374008717723
// MI455X (gfx1250) — compile-verified
//
#include <hip/hip_runtime.h>
#include <math.h>

#ifndef M_PI
#define M_PI 3.14159265358979323846
#endif

#define LDM  256   // padded leading dim for P, Q, and LU matrices
#define MAXN 256   // max frequencies handled (NFREQ == 256)
#define NB   12    // LU panel width (252 = 21*12)

typedef float v2f __attribute__((ext_vector_type(2)));
typedef float v8f __attribute__((ext_vector_type(8)));

// ---------------------------------------------------------------------------
// Per-frequency scalars:  u = i*omega
//   a = w - 0.5*om^2 - 0.5i*om
//   b = kappa + i*rho*om
//   gamma = b^2 - 2a        (coefficient of Q in denom)
// ---------------------------------------------------------------------------
__global__ void prep_scalars(const double* __restrict__ omega,
                             const double* __restrict__ w,
                             const double* __restrict__ kappa,
                             const double* __restrict__ rho,
                             double* __restrict__ scal, int B)
{
    int f = blockIdx.x * blockDim.x + threadIdx.x;
    if (f >= B) return;
    double om  = omega[f];
    double are = w[f] - 0.5 * om * om;
    double aim = -0.5 * om;
    double bre = kappa[0];
    double bim = rho[0] * om;
    double b2re = bre * bre - bim * bim;
    double b2im = 2.0 * bre * bim;
    double* s = scal + (size_t)f * 8;
    s[0] = are; s[1] = aim;
    s[2] = bre; s[3] = bim;
    s[4] = b2re - 2.0 * are;   // gamma_re
    s[5] = b2im - 2.0 * aim;   // gamma_im
}

// ---------------------------------------------------------------------------
// P = KK + KK^T (f64 VALU), Q = KK * KK^T via V_WMMA_F32_16X16X4_F32 with
// double-single (hi/lo) splitting; f64 accumulation flushed every K-chunk.
// One wave (32 lanes) per 16x16 output tile.
// ---------------------------------------------------------------------------
__global__ void pq_build(const double* __restrict__ KK,
                         double* __restrict__ P,
                         double* __restrict__ Q, int n)
{
    int row0 = blockIdx.x * 16;
    int col0 = blockIdx.y * 16;
    int lane = threadIdx.x;                 // 0..31 (wave32)
    int M    = lane & 15;                   // also N for the B operand
    int kk   = (lane < 16) ? 0 : 2;

    double accd[8];
#pragma unroll
    for (int v = 0; v < 8; ++v) accd[v] = 0.0;

    int nk = (n + 3) >> 2;                  // K chunks of 4 (252 = 63*4)
    for (int kc = 0; kc < nk; ++kc) {
        int kb = kc * 4 + kk;
        double a0 = 0.0, a1 = 0.0, b0 = 0.0, b1 = 0.0;
        if (row0 + M < n) {                 // A[M][k] = KK[row0+M][k]
            const double* ra = KK + (size_t)(row0 + M) * n;
            if (kb     < n) a0 = ra[kb];
            if (kb + 1 < n) a1 = ra[kb + 1];
        }
        if (col0 + M < n) {                 // B[k][N] = KK^T[k][col0+N] = KK[col0+N][k]
            const double* rb = KK + (size_t)(col0 + M) * n;
            if (kb     < n) b0 = rb[kb];
            if (kb + 1 < n) b1 = rb[kb + 1];
        }
        float a0h = (float)a0, a1h = (float)a1;
        float b0h = (float)b0, b1h = (float)b1;
        float a0l = (float)(a0 - (double)a0h), a1l = (float)(a1 - (double)a1h);
        float b0l = (float)(b0 - (double)b0h), b1l = (float)(b1 - (double)b1h);
        v2f Ah = {a0h, a1h}, Al = {a0l, a1l};
        v2f Bh = {b0h, b1h}, Bl = {b0l, b1l};
        v8f chh = {}, chl = {}, clh = {};
        chh = __builtin_amdgcn_wmma_f32_16x16x4_f32(false, Ah, false, Bh,
                                                    (short)0, chh, false, false);
        chl = __builtin_amdgcn_wmma_f32_16x16x4_f32(false, Ah, false, Bl,
                                                    (short)0, chl, false, false);
        clh = __builtin_amdgcn_wmma_f32_16x16x4_f32(false, Al, false, Bh,
                                                    (short)0, clh, false, false);
#pragma unroll
        for (int v = 0; v < 8; ++v)
            accd[v] += (double)chh[v] + (double)chl[v] + (double)clh[v];
    }

    // C/D layout: VGPR v -> M = v (lanes 0-15) or v+8 (lanes 16-31); N = lane&15
    int N     = lane & 15;
    int mbase = (lane < 16) ? 0 : 8;
#pragma unroll
    for (int v = 0; v < 8; ++v) {
        int Mi = row0 + mbase + v;
        int Nj = col0 + N;
        if (Mi < n && Nj < n) {
            Q[(size_t)Mi * LDM + Nj] = accd[v];
            P[(size_t)Mi * LDM + Nj] =
                KK[(size_t)Mi * n + Nj] + KK[(size_t)Nj * n + Mi];
        }
    }
}

// ---------------------------------------------------------------------------
// Per-frequency: phase 0 builds X = I - b*KK, blocked pivoted complex LU ->
// logdet(X). Phase 1 builds denom = I - b*P + gamma*Q in the same slot,
// blocked LU -> logdet(denom), solves denom*y = g, quad = a*dt*g^T y.
// One block (256 threads = 8 wave32s) per frequency. Blocked (NB) right-
// looking LU: panel columns updated eagerly (cache-resident slab), trailing
// matrix updated once per panel by a register-blocked rank-NB GEMM.
// ---------------------------------------------------------------------------
__global__ void lu_kernel(const double* __restrict__ KK,
                          const double* __restrict__ P,
                          const double* __restrict__ Q,
                          const double* __restrict__ scal,
                          const double* __restrict__ gvec,
                          const double* __restrict__ Tptr,
                          double* __restrict__ res,
                          double* __restrict__ mats,
                          int base, int n, int B)
{
    int f = base + blockIdx.x;
    if (f >= B) return;

    double* MR = mats + (size_t)blockIdx.x * (2 * (size_t)LDM * LDM);
    double* MI = MR + (size_t)LDM * LDM;

    const double* s = scal + (size_t)f * 8;
    double are = s[0], aim = s[1], bre = s[2], bim = s[3], gre = s[4], gim = s[5];

    int tid  = threadIdx.x;
    int lane = tid & 31, wid = tid >> 5;

    __shared__ double sGr[MAXN], sGi[MAXN];
    __shared__ double sRedV[8], sRedV2[8];
    __shared__ int    sRedI[8];
    __shared__ int    sPiv;

    for (int phase = 0; phase < 2; ++phase) {
        // ---- build matrix ----
        for (int i = 0; i < n; ++i) {
            for (int j = tid; j < n; j += blockDim.x) {
                double re, im;
                if (phase == 0) {
                    double k = KK[(size_t)i * n + j];
                    re = -bre * k; im = -bim * k;
                } else {
                    double p = P[(size_t)i * LDM + j];
                    double q = Q[(size_t)i * LDM + j];
                    re = -bre * p + gre * q;
                    im = -bim * p + gim * q;
                }
                if (i == j) re += 1.0;
                MR[(size_t)i * LDM + j] = re;
                MI[(size_t)i * LDM + j] = im;
            }
        }
        if (phase == 1)
            for (int j = tid; j < n; j += blockDim.x) { sGr[j] = gvec[j]; sGi[j] = 0.0; }
        __syncthreads();

        double ldre = 0.0, ldim = 0.0;  // thread 0 accumulates

        // ---- blocked partial-pivot LU ----
        for (int k0 = 0; k0 < n; k0 += NB) {
            int kend = (k0 + NB < n) ? (k0 + NB) : n;

            // --- panel factorization: columns k0..kend-1 ---
            for (int k = k0; k < kend; ++k) {
                // pivot search over rows k..n-1 of column k (cabs1)
                double bv = -1.0; int bi = -1;
                int i = k + tid;
                if (i < n) {
                    bv = fabs(MR[(size_t)i * LDM + k]) + fabs(MI[(size_t)i * LDM + k]);
                    bi = i;
                }
#pragma unroll
                for (int off = 16; off > 0; off >>= 1) {
                    double ov = __shfl_xor(bv, off, 32);
                    int    oi = __shfl_xor(bi, off, 32);
                    if (ov > bv) { bv = ov; bi = oi; }
                }
                if (lane == 0) { sRedV[wid] = bv; sRedI[wid] = bi; }
                __syncthreads();
                if (tid == 0) {
                    double mv = sRedV[0]; int mi = sRedI[0];
                    for (int t = 1; t < 8; ++t)
                        if (sRedV[t] > mv) { mv = sRedV[t]; mi = sRedI[t]; }
                    sPiv = mi;
                }
                __syncthreads();
                int p = sPiv;

                if (p != k) {            // full-row swap
                    for (int j = tid; j < n; j += blockDim.x) {
                        double t1 = MR[(size_t)k * LDM + j];
                        MR[(size_t)k * LDM + j] = MR[(size_t)p * LDM + j];
                        MR[(size_t)p * LDM + j] = t1;
                        double t2 = MI[(size_t)k * LDM + j];
                        MI[(size_t)k * LDM + j] = MI[(size_t)p * LDM + j];
                        MI[(size_t)p * LDM + j] = t2;
                    }
                    if (tid == 0 && phase == 1) {
                        double t = sGr[k]; sGr[k] = sGr[p]; sGr[p] = t;
                        t = sGi[k]; sGi[k] = sGi[p]; sGi[p] = t;
                    }
                }
                __syncthreads();

                double pr = MR[(size_t)k * LDM + k], pi = MI[(size_t)k * LDM + k];
                double d  = pr * pr + pi * pi;
                if (tid == 0) {
                    ldre += 0.5 * log(d);
                    ldim += atan2(pi, pr);
                    if (p != k) ldim += M_PI;     // row swap flips det sign
                }
                double ir = pr / d, ii = -pi / d;

                // scale column k and eagerly update remaining panel columns
                for (int i2 = k + 1 + tid; i2 < n; i2 += blockDim.x) {
                    double vr = MR[(size_t)i2 * LDM + k], vi = MI[(size_t)i2 * LDM + k];
                    double lr = vr * ir - vi * ii;
                    double li = vr * ii + vi * ir;
                    MR[(size_t)i2 * LDM + k] = lr;
                    MI[(size_t)i2 * LDM + k] = li;
                    for (int j = k + 1; j < kend; ++j) {
                        double ur = MR[(size_t)k * LDM + j];   // uniform
                        double ui = MI[(size_t)k * LDM + j];
                        MR[(size_t)i2 * LDM + j] -= lr * ur - li * ui;
                        MI[(size_t)i2 * LDM + j] -= lr * ui + li * ur;
                    }
                }
                __syncthreads();
            }

            // --- U12 = L11^{-1} * A12 (unit lower trsm, row by row) ---
            for (int r = k0 + 1; r < kend; ++r) {
                for (int j = kend + tid; j < n; j += blockDim.x) {
                    double ar = MR[(size_t)r * LDM + j], ai = MI[(size_t)r * LDM + j];
                    for (int m = k0; m < r; ++m) {
                        double lr = MR[(size_t)r * LDM + m];   // uniform
                        double li = MI[(size_t)r * LDM + m];
                        double ur = MR[(size_t)m * LDM + j];
                        double ui = MI[(size_t)m * LDM + j];
                        ar -= lr * ur - li * ui;
                        ai -= lr * ui + li * ur;
                    }
                    MR[(size_t)r * LDM + j] = ar;
                    MI[(size_t)r * LDM + j] = ai;
                }
                __syncthreads();
            }

            // --- trailing GEMM: A22 -= L21 * U12 (register-blocked rank-NB) ---
            int nb = kend - k0;
            for (int j = kend + tid; j < n; j += blockDim.x) {
                double ur[NB], ui[NB];
#pragma unroll
                for (int m = 0; m < NB; ++m) {
                    if (m < nb) {
                        ur[m] = MR[(size_t)(k0 + m) * LDM + j];
                        ui[m] = MI[(size_t)(k0 + m) * LDM + j];
                    } else { ur[m] = 0.0; ui[m] = 0.0; }
                }
                for (int i2 = kend; i2 < n; ++i2) {
                    if (i2 + 4 < n) {   // prefetch C-row stream (global_prefetch_b8)
                        __builtin_prefetch(&MR[(size_t)(i2 + 4) * LDM + j]);
                        __builtin_prefetch(&MI[(size_t)(i2 + 4) * LDM + j]);
                    }
                    double cr = MR[(size_t)i2 * LDM + j], ci = MI[(size_t)i2 * LDM + j];
#pragma unroll
                    for (int m = 0; m < NB; ++m) {
                        if (m < nb) {
                            double lr = MR[(size_t)i2 * LDM + k0 + m];  // uniform
                            double li = MI[(size_t)i2 * LDM + k0 + m];
                            cr -= lr * ur[m] - li * ui[m];
                            ci -= lr * ui[m] + li * ur[m];
                        }
                    }
                    MR[(size_t)i2 * LDM + j] = cr;
                    MI[(size_t)i2 * LDM + j] = ci;
                }
            }
            __syncthreads();
        }

        if (tid == 0) {
            if (phase == 0) { res[f * 6 + 0] = ldre; res[f * 6 + 1] = ldim; }
            else            { res[f * 6 + 2] = ldre; res[f * 6 + 3] = ldim; }
        }

        // ---- solve denom * y = g and quad = a*dt*g^T y ----
        if (phase == 1) {
            for (int k = 0; k < n; ++k) {          // forward (unit lower)
                double ykr = sGr[k], yki = sGi[k];
                for (int i2 = k + 1 + tid; i2 < n; i2 += blockDim.x) {
                    double lr = MR[(size_t)i2 * LDM + k], li = MI[(size_t)i2 * LDM + k];
                    sGr[i2] -= lr * ykr - li * yki;
                    sGi[i2] -= lr * yki + li * ykr;
                }
                __syncthreads();
            }
            for (int k = n - 1; k >= 0; --k) {     // backward
                if (tid == 0) {
                    double ur = MR[(size_t)k * LDM + k], ui = MI[(size_t)k * LDM + k];
                    double dd = ur * ur + ui * ui;
                    double xr = (sGr[k] * ur + sGi[k] * ui) / dd;
                    double xi = (sGi[k] * ur - sGr[k] * ui) / dd;
                    sGr[k] = xr; sGi[k] = xi;
                }
                __syncthreads();
                double xkr = sGr[k], xki = sGi[k];
                for (int i2 = tid; i2 < k; i2 += blockDim.x) {
                    double ur = MR[(size_t)i2 * LDM + k], ui = MI[(size_t)i2 * LDM + k];
                    sGr[i2] -= ur * xkr - ui * xki;
                    sGi[i2] -= ur * xki + ui * xkr;
                }
                __syncthreads();
            }
            double pr2 = 0.0, pi2 = 0.0;
            for (int j = tid; j < n; j += blockDim.x) {
                double gj = gvec[j];
                pr2 += gj * sGr[j]; pi2 += gj * sGi[j];
            }
#pragma unroll
            for (int off = 16; off > 0; off >>= 1) {
                pr2 += __shfl_xor(pr2, off, 32);
                pi2 += __shfl_xor(pi2, off, 32);
            }
            if (lane == 0) { sRedV[wid] = pr2; sRedV2[wid] = pi2; }
            __syncthreads();
            if (tid == 0) {
                double sr = 0.0, si = 0.0;
                for (int t = 0; t < 8; ++t) { sr += sRedV[t]; si += sRedV2[t]; }
                double dt = Tptr[0] / (double)n;
                res[f * 6 + 4] = dt * (are * sr - aim * si);
                res[f * 6 + 5] = dt * (are * si + aim * sr);
            }
        }
        __syncthreads();
    }
}

// ---------------------------------------------------------------------------
// values = exp(u*(ln m + r*T) + quad - 0.5*logdet(D)), logdet(D) = Ld - 2*Lx
// with Im wrapped to (-pi,pi] (matches angle(det)+principal sqrt), then the
// sequential branch-cut sign scan.
// ---------------------------------------------------------------------------
__global__ void finalize_kernel(const double* __restrict__ omega,
                                const double* __restrict__ rr,
                                const double* __restrict__ TT,
                                const double* __restrict__ mm,
                                const double* __restrict__ res,
                                float* __restrict__ out, int B, int out_size)
{
    __shared__ double sTh[MAXN];
    __shared__ float  sSg[MAXN];
    int f = threadIdx.x;
    double vre = 0.0, vim = 0.0;
    if (f < B) {
        double LxR = res[f * 6 + 0], LxI = res[f * 6 + 1];
        double LdR = res[f * 6 + 2], LdI = res[f * 6 + 3];
        double qr  = res[f * 6 + 4], qi  = res[f * 6 + 5];
        double r2 = LdR - 2.0 * LxR;
        double t  = LdI - 2.0 * LxI;
        const double TWO_PI = 6.283185307179586476925287;
        double th = t - TWO_PI * floor(t / TWO_PI + 0.5);   // (-pi, pi]
        double c0 = log(mm[0]) + rr[0] * TT[0];
        double om = omega[f];
        double Ere = qr - 0.5 * r2;
        double Eim = qi + om * c0 - 0.5 * th;
        double mag = exp(Ere);
        vre = mag * cos(Eim);
        vim = mag * sin(Eim);
        sTh[f] = th;
    }
    __syncthreads();
    if (threadIdx.x == 0) {
        int par = 0;
        sSg[0] = 1.0f;
        for (int i = 1; i < B; ++i) {
            if (fabs(sTh[i] - sTh[i - 1]) > 5.0) par ^= 1;
            sSg[i] = par ? -1.0f : 1.0f;
        }
    }
    __syncthreads();
    if (f < B) {
        float sg = sSg[f];
        if (2 * f     < out_size) out[2 * f]     = (float)vre * sg;
        if (2 * f + 1 < out_size) out[2 * f + 1] = (float)vim * sg;
    }
}

// ---------------------------------------------------------------------------
extern "C" void kernel_launch(void* const* d_in, const int* in_sizes, int n_in,
                              void* d_out, int out_size, void* d_ws, size_t ws_size,
                              hipStream_t stream)
{
    const double* omega = (const double*)d_in[0];
    const double* w     = (const double*)d_in[1];
    const double* KK    = (const double*)d_in[2];
    const double* g     = (const double*)d_in[3];
    const double* kappa = (const double*)d_in[4];
    const double* rho   = (const double*)d_in[5];
    const double* r     = (const double*)d_in[6];
    const double* T     = (const double*)d_in[7];
    const double* money = (const double*)d_in[8];

    int B = in_sizes[0];        // NFREQ (256)
    int n = in_sizes[3];        // NSTEP (252)
    if (B > MAXN) B = MAXN;

    double* wsd = (double*)d_ws;
    size_t off = 0;
    double* P    = wsd + off; off += (size_t)LDM * LDM;
    double* Q    = wsd + off; off += (size_t)LDM * LDM;
    double* scal = wsd + off; off += (size_t)MAXN * 8;
    double* res  = wsd + off; off += (size_t)MAXN * 6;
    double* mats = wsd + off;

    size_t slotDbl  = (size_t)2 * LDM * LDM;             // ~1 MB per frequency
    size_t availDbl = (ws_size / 8 > off) ? (ws_size / 8 - off) : 0;
    int slots = (int)(availDbl / slotDbl);
    if (slots < 1) slots = 1;
    if (slots > B) slots = B;

    prep_scalars<<<dim3((B + 255) / 256), dim3(256), 0, stream>>>(omega, w, kappa, rho, scal, B);

    int nt = (n + 15) / 16;
    pq_build<<<dim3(nt, nt), dim3(32), 0, stream>>>(KK, P, Q, n);

    for (int base = 0; base < B; base += slots) {
        int cnt = (B - base < slots) ? (B - base) : slots;
        lu_kernel<<<dim3(cnt), dim3(256), 0, stream>>>(KK, P, Q, scal, g, T, res, mats, base, n, B);
    }

    finalize_kernel<<<dim3(1), dim3(256), 0, stream>>>(omega, r, T, money, res, (float*)d_out, B, out_size);
}